// Net_40836549050800
// MI455X (gfx1250) — compile-verified
//
#include <hip/hip_runtime.h>
#include <hip/hip_bf16.h>
#include <math.h>

typedef __attribute__((ext_vector_type(2))) float v2f;
typedef __attribute__((ext_vector_type(8))) float v8f;

#define BATCH 4
#define SEQ   2048
#define EMBD  300
#define CCH   128
#define NTOK  1505   // stage-1 tokens actually needed (j = 0..1504)
#define T2    1500   // second maxpool keeps only t in [0,1500)
#define NTILE 94     // 94 tiles of 16 t-columns cover [0,1504)

// ---------------------------------------------------------------------------
// Zero the pooled2 accumulator (atomicMax on non-negative float bit patterns
// requires a 0x00000000 init; d_ws is poisoned with 0xAA).
// ---------------------------------------------------------------------------
__global__ void k_init(float* pooled2) {
    int i = threadIdx.x;
    if (i < BATCH * CCH) pooled2[i] = 0.0f;
}

// ---------------------------------------------------------------------------
// Stage 1: fused embedding gather + conv1(K=2,Cin=1) + ReLU + maxpool(300).
// One block per (batch, token). Pool window j == embedding row j plus the
// first element of row j+1. max_k relu(y_k) == max(0, max_k y_k).
// ---------------------------------------------------------------------------
__global__ __launch_bounds__(128) void k_stage1(const int* __restrict__ x,
                                                const float* __restrict__ emb,
                                                const float* __restrict__ w1,
                                                const float* __restrict__ b1,
                                                float* __restrict__ P1) {
    const int blk = blockIdx.x;
    const int b = blk / NTOK;
    const int j = blk % NTOK;
    const int tid = threadIdx.x;

    __shared__ float sv[EMBD + 1];

    const int tok0 = x[b * SEQ + j];
    const int tok1 = x[b * SEQ + j + 1];           // j+1 <= 1505 < 2048: safe
    for (int k = tid; k < EMBD; k += 128)
        sv[k] = emb[(long long)tok0 * EMBD + k];
    if (tid == 0) sv[EMBD] = emb[(long long)tok1 * EMBD];
    __syncthreads();

    const float wa   = w1[tid * 2 + 0];
    const float wb   = w1[tid * 2 + 1];
    const float bias = b1[tid];
    float m = 0.0f;                                 // relu floor
    #pragma unroll 4
    for (int k = 0; k < EMBD; ++k) {
        float y = fmaf(wa, sv[k], fmaf(wb, sv[k + 1], bias));
        m = fmaxf(m, y);
    }
    P1[((b * NTOK) + j) * CCH + tid] = m;           // layout [b][t][c]
}

// ---------------------------------------------------------------------------
// Stage 2: conv2 (128x128 matmul per tap per t-column) via V_WMMA_F32_16X16X4,
// fused with ReLU + maxpool(1500) using a lane-shuffle max reduce + atomicMax.
// Block = 256 threads = 8 waves; wave w owns cout tile [16w, 16w+16).
// ---------------------------------------------------------------------------
__global__ __launch_bounds__(256) void k_stage2(const float* __restrict__ P1,
                                                const float* __restrict__ w2,
                                                const float* __restrict__ b2,
                                                float* __restrict__ pooled2) {
    const int b    = blockIdx.x / NTILE;
    const int tile = blockIdx.x % NTILE;
    const int t0   = tile * 16;
    const int tid  = threadIdx.x;

    __shared__ float sP1[17 * CCH];                 // 17 t-columns x 128 cin

    for (int idx = tid; idx < 17 * CCH; idx += 256) {
        int col = idx >> 7, c = idx & 127;
        sP1[idx] = P1[((b * NTOK) + t0 + col) * CCH + c];   // t0+16 <= 1504 ok
    }
    __syncthreads();

    const int wave  = tid >> 5;
    const int lane  = tid & 31;
    const int n     = lane & 15;                    // N index; also M for A
    const int khalf = (lane < 16) ? 0 : 2;          // K half per ISA layout
    const int cout0 = wave * 16;

    v8f acc = {};
    // w2 flat layout [co][ci][tap]; one float4 per chunk carries both taps:
    // (a0.x, a1.x, a0.y, a1.y) at flat ((co*128 + ci)*2).
    const float* w2row = w2 + (long long)(cout0 + n) * CCH * 2;
    for (int ci0 = 0; ci0 < CCH; ci0 += 4) {
        float4 aq = *(const float4*)(w2row + (ci0 + khalf) * 2);
        v2f a0 = { aq.x, aq.z };                    // tap 0: K=khalf, khalf+1
        v2f a1 = { aq.y, aq.w };                    // tap 1
        v2f bb0 = *(const v2f*)&sP1[(n + 0) * CCH + ci0 + khalf];
        v2f bb1 = *(const v2f*)&sP1[(n + 1) * CCH + ci0 + khalf];
        acc = __builtin_amdgcn_wmma_f32_16x16x4_f32(false, a0, false, bb0,
                                                    (short)0, acc, false, false);
        acc = __builtin_amdgcn_wmma_f32_16x16x4_f32(false, a1, false, bb1,
                                                    (short)0, acc, false, false);
    }

    const bool tvalid = (t0 + n) < T2;              // mask pool-window overrun
    #pragma unroll
    for (int r = 0; r < 8; ++r) {
        int row = cout0 + r + ((lane < 16) ? 0 : 8);
        float v = fmaxf(acc[r] + b2[row], 0.0f);
        if (!tvalid) v = 0.0f;                      // neutral for max of relu
        // max over the 16 lanes (16 t-columns) sharing this output row
        for (int off = 1; off < 16; off <<= 1)
            v = fmaxf(v, __shfl_xor(v, off, 16));
        if ((lane & 15) == 0)
            atomicMax((unsigned int*)&pooled2[b * CCH + row],
                      __float_as_uint(v));          // valid: v >= 0
    }
}

// ---------------------------------------------------------------------------
// Head: [4,128] @ wl.T + bl, softmax. One wave.
// ---------------------------------------------------------------------------
__global__ void k_head(const float* __restrict__ pooled2,
                       const float* __restrict__ wl,
                       const float* __restrict__ bl,
                       float* __restrict__ out) {
    __shared__ float lg[BATCH * 5];
    const int t = threadIdx.x;
    if (t < BATCH * 5) {
        int b = t / 5, o = t % 5;
        float s = bl[o];
        for (int c = 0; c < CCH; ++c)
            s = fmaf(pooled2[b * CCH + c], wl[o * CCH + c], s);
        lg[t] = s;
    }
    __syncthreads();
    if (t < BATCH * 5) {
        int b = t / 5;
        float mx = lg[b * 5];
        for (int o = 1; o < 5; ++o) mx = fmaxf(mx, lg[b * 5 + o]);
        float sum = 0.0f;
        for (int o = 0; o < 5; ++o) sum += expf(lg[b * 5 + o] - mx);
        out[t] = expf(lg[t] - mx) / sum;
    }
}

extern "C" void kernel_launch(void* const* d_in, const int* in_sizes, int n_in,
                              void* d_out, int out_size, void* d_ws, size_t ws_size,
                              hipStream_t stream) {
    const int*   x   = (const int*)d_in[0];
    const float* emb = (const float*)d_in[1];
    const float* w1  = (const float*)d_in[2];
    const float* b1  = (const float*)d_in[3];
    const float* w2  = (const float*)d_in[4];
    const float* b2  = (const float*)d_in[5];
    const float* wl  = (const float*)d_in[6];
    const float* bl  = (const float*)d_in[7];

    float* pooled2 = (float*)d_ws;                  // 512 floats
    float* P1      = pooled2 + BATCH * CCH;         // 4*1505*128 floats

    k_init  <<<1, 512, 0, stream>>>(pooled2);
    k_stage1<<<BATCH * NTOK, 128, 0, stream>>>(x, emb, w1, b1, P1);
    k_stage2<<<BATCH * NTILE, 256, 0, stream>>>(P1, w2, b2, pooled2);
    k_head  <<<1, 32, 0, stream>>>(pooled2, wl, bl, (float*)d_out);
}